// LinearAttn_37752762532670
// MI455X (gfx1250) — compile-verified
//
#include <hip/hip_runtime.h>

typedef __attribute__((ext_vector_type(2))) float v2f;
typedef __attribute__((ext_vector_type(8))) float v8f;
typedef __attribute__((vector_size(16))) int v4i_t;

#define NN   50000
#define NE   800000
#define DIM  64
#define COLS 192
#define EPS  1e-5f

#if __has_builtin(__builtin_amdgcn_global_load_async_to_lds_b128) && \
    __has_builtin(__builtin_amdgcn_s_wait_asynccnt)
#define HAVE_ASYNC_LDS 1
#else
#define HAVE_ASYNC_LDS 0
#endif

// Cooperative global->LDS tile copy (rows x 64 floats, LDS stride 68).
// Uses gfx1250 async-to-LDS path (ASYNCcnt) when available.
__device__ __forceinline__ void load_tile(
    const float* __restrict__ g, float (*lds)[68], int rows, int t, int nthreads)
{
    const int chunks = rows * (DIM / 4);        // 16-byte chunks
    for (int i = t; i < chunks; i += nthreads) {
        const int r = i >> 4;
        const int c = (i & 15) << 2;
#if HAVE_ASYNC_LDS
        __builtin_amdgcn_global_load_async_to_lds_b128(
            (__attribute__((address_space(1))) v4i_t*)(g + (size_t)r * DIM + c),
            (__attribute__((address_space(3))) v4i_t*)&lds[r][c],
            0, 0);
#else
        *(float4*)&lds[r][c] = *(const float4*)(g + (size_t)r * DIM + c);
#endif
    }
}

__device__ __forceinline__ void wait_tile_ready()
{
#if HAVE_ASYNC_LDS
    __builtin_amdgcn_s_wait_asynccnt(0);
#endif
    __syncthreads();
}

// ---------------------------------------------------------------------------
// Kernel A: LayerNorm + QKV projection, fp32 WMMA (V_WMMA_F32_16X16X4_F32)
// 16 nodes per block, 192 threads = 6 waves, each wave does 2 col-tiles.
// ---------------------------------------------------------------------------
__global__ __launch_bounds__(192) void ln_qkv_kernel(
    const float* __restrict__ node_feat,
    const float* __restrict__ W_qkv,
    const float* __restrict__ b_qkv,
    const float* __restrict__ ln_g,
    const float* __restrict__ ln_b,
    float* __restrict__ qkv_out)
{
    __shared__ float xs[16][68];               // padded stride: avoid bank conflicts
    const int t = threadIdx.x;
    const int node0 = blockIdx.x * 16;

    load_tile(node_feat + (size_t)node0 * DIM, xs, 16, t, 192);
    wait_tile_ready();

    if (t < 16) {                              // per-row layernorm
        float mu = 0.f;
        for (int c = 0; c < DIM; ++c) mu += xs[t][c];
        mu *= (1.f / DIM);
        float var = 0.f;
        for (int c = 0; c < DIM; ++c) { float d = xs[t][c] - mu; var += d * d; }
        var *= (1.f / DIM);
        float rs = rsqrtf(var + EPS);
        for (int c = 0; c < DIM; ++c)
            xs[t][c] = (xs[t][c] - mu) * rs * ln_g[c] + ln_b[c];
    }
    __syncthreads();

    const int wave = t >> 5;
    const int lane = t & 31;
    const int m    = lane & 15;
    const int ksel = (lane >> 4) << 1;         // 0 for lanes 0-15, 2 for 16-31

    for (int ct = wave; ct < 12; ct += 6) {
        const int c0 = ct * 16;
        v8f acc = {};
        #pragma unroll
        for (int kk = 0; kk < 16; ++kk) {      // K = 64 as 16 x (K=4) WMMAs
            const int k0 = kk * 4 + ksel;
            v2f a, b;
            a.x = xs[m][k0];
            a.y = xs[m][k0 + 1];
            b.x = W_qkv[(size_t)k0 * COLS + c0 + m];
            b.y = W_qkv[(size_t)(k0 + 1) * COLS + c0 + m];
            acc = __builtin_amdgcn_wmma_f32_16x16x4_f32(
                false, a, false, b, (short)0, acc, false, false);
        }
        const int rbase = (lane >> 4) << 3;
        const int colc  = c0 + m;
        const float bias = b_qkv[colc];
        #pragma unroll
        for (int v = 0; v < 8; ++v)
            qkv_out[(size_t)(node0 + rbase + v) * COLS + colc] = acc[v] + bias;
    }
}

// ---------------------------------------------------------------------------
// Kernel B: input_vec = node_vec @ W_vec  (48 rows x 192 cols per block),
// then input_dot = sum over the 3-axis of in1*in2, and input_3 stored.
// ---------------------------------------------------------------------------
__global__ __launch_bounds__(192) void vec_gemm_kernel(
    const float* __restrict__ node_vec,
    const float* __restrict__ W_vec,
    float* __restrict__ input_dot,
    float* __restrict__ input3)
{
    __shared__ float xa[48][68];
    __shared__ float ov[48][COLS];
    const int t = threadIdx.x;
    const int node0 = blockIdx.x * 16;
    const int row0  = node0 * 3;

    load_tile(node_vec + (size_t)row0 * DIM, xa, 48, t, 192);
    wait_tile_ready();

    const int wave = t >> 5;
    const int lane = t & 31;
    const int m    = lane & 15;
    const int ksel = (lane >> 4) << 1;

    for (int tile = wave; tile < 36; tile += 6) {   // 3 row-tiles x 12 col-tiles
        const int r0 = (tile % 3) * 16;
        const int c0 = (tile / 3) * 16;
        v8f acc = {};
        #pragma unroll
        for (int kk = 0; kk < 16; ++kk) {
            const int k0 = kk * 4 + ksel;
            v2f a, b;
            a.x = xa[r0 + m][k0];
            a.y = xa[r0 + m][k0 + 1];
            b.x = W_vec[(size_t)k0 * COLS + c0 + m];
            b.y = W_vec[(size_t)(k0 + 1) * COLS + c0 + m];
            acc = __builtin_amdgcn_wmma_f32_16x16x4_f32(
                false, a, false, b, (short)0, acc, false, false);
        }
        const int rbase = r0 + ((lane >> 4) << 3);
        const int colc  = c0 + m;
        #pragma unroll
        for (int v = 0; v < 8; ++v)
            ov[rbase + v][colc] = acc[v];
    }
    __syncthreads();

    for (int i = t; i < 16 * DIM; i += 192) {       // input_dot
        int nl = i >> 6, d = i & 63;
        float s = 0.f;
        #pragma unroll
        for (int c = 0; c < 3; ++c)
            s += ov[nl * 3 + c][d] * ov[nl * 3 + c][DIM + d];
        input_dot[(size_t)(node0 + nl) * DIM + d] = s;
    }
    for (int i = t; i < 48 * DIM; i += 192) {       // input_3
        int r = i >> 6, d = i & 63;
        input3[(size_t)(row0 + r) * DIM + d] = ov[r][2 * DIM + d];
    }
}

// ---------------------------------------------------------------------------
// Kernel C: per-edge attention + scatter-add. One wave32 per edge; lane l
// owns flat dims l (heads 0-7) and l+32 (heads 8-15). Per-head (4-dim) dot
// via two shfl_xor rounds. Exact gelu. f32 global atomics for segment_sum.
// ---------------------------------------------------------------------------
__device__ __forceinline__ float gelu_exact(float x) {
    return 0.5f * x * (1.f + erff(x * 0.70710678118654752f));
}

__global__ __launch_bounds__(256) void edge_kernel(
    const float* __restrict__ qkv,
    const float* __restrict__ edge_feat,
    const float* __restrict__ edge_vec,
    const float* __restrict__ radial,
    const int* __restrict__ row,
    const int* __restrict__ col,
    float* __restrict__ m_feat,
    float* __restrict__ m_vec)
{
    const int wave = threadIdx.x >> 5;
    const int lane = threadIdx.x & 31;
    const int e = blockIdx.x * 8 + wave;
    if (e >= NE) return;

    // warm the edge_feat row (used after the dot/gelu dependency chain)
    __builtin_prefetch(&edge_feat[(size_t)e * DIM + lane * 2], 0, 1);

    const int r = row[e];
    const int c = col[e];
    const float rad = radial[e];
    const float ev0 = edge_vec[(size_t)e * 3 + 0];
    const float ev1 = edge_vec[(size_t)e * 3 + 1];
    const float ev2 = edge_vec[(size_t)e * 3 + 2];

    // flat dim i = h*4+d -> qkv row cols: q = h*12+d, k = +4, v = +8
    const int h  = lane >> 2;
    const int d  = lane & 3;
    const int cq = h * 12 + d;
    const size_t qb = (size_t)r * COLS;
    const size_t kb = (size_t)c * COLS;

    float p0 = qkv[qb + cq]      * qkv[kb + cq + 4];    // head h
    float p1 = qkv[qb + cq + 96] * qkv[kb + cq + 100];  // head h+8
    p0 += __shfl_xor(p0, 1, 32);  p0 += __shfl_xor(p0, 2, 32);
    p1 += __shfl_xor(p1, 1, 32);  p1 += __shfl_xor(p1, 2, 32);
    const float attn0 = gelu_exact(p0) * rad;
    const float attn1 = gelu_exact(p1) * rad;

    const float v0 = qkv[kb + cq + 8];
    const float v1 = qkv[kb + cq + 104];
    const int i0 = lane, i1 = lane + 32;
    const float ef0 = edge_feat[(size_t)e * DIM + i0];
    const float ef1 = edge_feat[(size_t)e * DIM + i1];

    const size_t fb = (size_t)r * DIM;
    atomicAdd(&m_feat[fb + i0], v0 * ef0 * attn0);
    atomicAdd(&m_feat[fb + i1], v1 * ef1 * attn1);
    const size_t vb = (size_t)r * 3 * DIM;
    atomicAdd(&m_vec[vb + 0 * DIM + i0], v0 * ev0);
    atomicAdd(&m_vec[vb + 0 * DIM + i1], v1 * ev0);
    atomicAdd(&m_vec[vb + 1 * DIM + i0], v0 * ev1);
    atomicAdd(&m_vec[vb + 1 * DIM + i1], v1 * ev1);
    atomicAdd(&m_vec[vb + 2 * DIM + i0], v0 * ev2);
    atomicAdd(&m_vec[vb + 2 * DIM + i1], v1 * ev2);
}

// ---------------------------------------------------------------------------
// Kernel D: out = m_feat @ W_out + b_out (WMMA), then final combines.
// ---------------------------------------------------------------------------
__global__ __launch_bounds__(192) void out_kernel(
    const float* __restrict__ m_feat,
    const float* __restrict__ W_out,
    const float* __restrict__ b_out,
    const float* __restrict__ input_dot,
    const float* __restrict__ input3,
    const float* __restrict__ m_vec,
    float* __restrict__ dnf,          // delta_node_feat  (N,64)
    float* __restrict__ dnv)          // delta_node_vec   (N,3,64)
{
    __shared__ float ma[16][68];
    __shared__ float ol[16][COLS];
    const int t = threadIdx.x;
    const int node0 = blockIdx.x * 16;

    load_tile(m_feat + (size_t)node0 * DIM, ma, 16, t, 192);
    wait_tile_ready();

    const int wave = t >> 5;
    const int lane = t & 31;
    const int m    = lane & 15;
    const int ksel = (lane >> 4) << 1;

    for (int ct = wave; ct < 12; ct += 6) {
        const int c0 = ct * 16;
        v8f acc = {};
        #pragma unroll
        for (int kk = 0; kk < 16; ++kk) {
            const int k0 = kk * 4 + ksel;
            v2f a, b;
            a.x = ma[m][k0];
            a.y = ma[m][k0 + 1];
            b.x = W_out[(size_t)k0 * COLS + c0 + m];
            b.y = W_out[(size_t)(k0 + 1) * COLS + c0 + m];
            acc = __builtin_amdgcn_wmma_f32_16x16x4_f32(
                false, a, false, b, (short)0, acc, false, false);
        }
        const int rbase = (lane >> 4) << 3;
        const int colc  = c0 + m;
        const float bias = b_out[colc];
        #pragma unroll
        for (int v = 0; v < 8; ++v)
            ol[rbase + v][colc] = acc[v] + bias;
    }
    __syncthreads();

    for (int i = t; i < 16 * DIM; i += 192) {       // delta_node_feat
        int nl = i >> 6, d = i & 63;
        float o2 = ol[nl][DIM + d], o3 = ol[nl][2 * DIM + d];
        dnf[(size_t)(node0 + nl) * DIM + d] =
            input_dot[(size_t)(node0 + nl) * DIM + d] * o2 + o3;
    }
    for (int i = t; i < 48 * DIM; i += 192) {       // delta_node_vec
        int r = i >> 6, d = i & 63;
        float o1 = ol[r / 3][d];
        size_t ri = (size_t)(node0 * 3 + r) * DIM + d;
        dnv[ri] = input3[ri] * o1 + m_vec[ri];
    }
}

// ---------------------------------------------------------------------------
extern "C" void kernel_launch(void* const* d_in, const int* in_sizes, int n_in,
                              void* d_out, int out_size, void* d_ws, size_t ws_size,
                              hipStream_t stream)
{
    const float* node_feat = (const float*)d_in[0];
    const float* edge_feat = (const float*)d_in[1];
    const float* node_vec  = (const float*)d_in[2];
    const float* edge_vec  = (const float*)d_in[3];
    const float* radial    = (const float*)d_in[4];
    const float* W_qkv     = (const float*)d_in[5];
    const float* b_qkv     = (const float*)d_in[6];
    const float* W_out     = (const float*)d_in[7];
    const float* b_out     = (const float*)d_in[8];
    const float* W_vec     = (const float*)d_in[9];
    const float* ln_g      = (const float*)d_in[10];
    const float* ln_b      = (const float*)d_in[11];
    const int*   row       = (const int*)d_in[12];
    const int*   col       = (const int*)d_in[13];

    float* ws        = (float*)d_ws;
    float* qkv       = ws;                               // N*192
    float* input_dot = qkv + (size_t)NN * COLS;          // N*64
    float* input3    = input_dot + (size_t)NN * DIM;     // N*192
    float* m_feat    = input3 + (size_t)NN * 3 * DIM;    // N*64
    float* m_vec     = m_feat + (size_t)NN * DIM;        // N*192

    float* dnf = (float*)d_out;
    float* dnv = dnf + (size_t)NN * DIM;

    (void)hipMemsetAsync(m_feat, 0, (size_t)NN * DIM * sizeof(float), stream);
    (void)hipMemsetAsync(m_vec,  0, (size_t)NN * 3 * DIM * sizeof(float), stream);

    ln_qkv_kernel<<<NN / 16, 192, 0, stream>>>(node_feat, W_qkv, b_qkv, ln_g, ln_b, qkv);
    vec_gemm_kernel<<<NN / 16, 192, 0, stream>>>(node_vec, W_vec, input_dot, input3);
    edge_kernel<<<NE / 8, 256, 0, stream>>>(qkv, edge_feat, edge_vec, radial,
                                            row, col, m_feat, m_vec);
    out_kernel<<<NN / 16, 192, 0, stream>>>(m_feat, W_out, b_out, input_dot,
                                            input3, m_vec, dnf, dnv);
}